// GNNLayer_70231305225060
// MI455X (gfx1250) — compile-verified
//
#include <hip/hip_runtime.h>
#include <hip/hip_bf16.h>

typedef __attribute__((ext_vector_type(2))) float v2f;
typedef __attribute__((ext_vector_type(8))) float v8f;

#define IN_D 128
#define OUT_D 128

// ---------------- zero workspace (agg + deg) ----------------
__global__ void gnn_zero_kernel(float* __restrict__ p, long n) {
  long i = (long)blockIdx.x * blockDim.x + threadIdx.x;
  const long stride = (long)gridDim.x * blockDim.x;
  for (; i < n; i += stride) p[i] = 0.0f;
}

// ---------------- edge scatter: agg[dst] += x[src], deg[dst] += 1 ----------------
// One wave32 per edge; each lane handles 4 features (coalesced 128B per step).
__global__ void gnn_scatter_kernel(const float* __restrict__ x,
                                   const long long* __restrict__ ei,
                                   float* __restrict__ agg,
                                   float* __restrict__ deg,
                                   int nEdges) {
  const int t = blockIdx.x * blockDim.x + threadIdx.x;
  const int e = t >> 5;
  if (e >= nEdges) return;
  const int lane = t & 31;
  const long long src = ei[e];
  const long long dst = ei[(long long)nEdges + e];
  const float* xs = x + src * (long long)IN_D;
  float* ad = agg + dst * (long long)IN_D;
#pragma unroll
  for (int i = 0; i < IN_D / 32; ++i) {
    const int f = lane + i * 32;
    atomicAdd(&ad[f], xs[f]);
  }
  if (lane == 0) atomicAdd(&deg[dst], 1.0f);
}

// ---------------- fused dual GEMM + bias + relu via f32 WMMA ----------------
// D = relu( x @ Ws + (agg/deg) @ Wn + bs + bn ), 16x16 tile per wave.
__global__ __launch_bounds__(256) void gnn_gemm_kernel(
    const float* __restrict__ x, const float* __restrict__ agg,
    const float* __restrict__ deg, const float* __restrict__ Ws,
    const float* __restrict__ bsv, const float* __restrict__ Wn,
    const float* __restrict__ bnv, float* __restrict__ out, int nNodes) {
  const int m0 = blockIdx.x * 16;
  const int n0 = (threadIdx.x >> 5) * 16;   // 8 waves -> 8 N-tiles of 16
  const int lane = threadIdx.x & 31;
  const int hi = lane >> 4;                 // half-wave select
  const int lr = lane & 15;

  int arow = m0 + lr;
  if (arow >= nNodes) arow = nNodes - 1;    // clamp loads on tail tile (cndmask, no exec change)
  const int koff = hi * 2;                  // A/B k-offset per ISA frag layout

  const float rdeg = 1.0f / fmaxf(deg[arow], 1.0f);
  const float* xr = x   + (long)arow * IN_D + koff;
  const float* gr = agg + (long)arow * IN_D + koff;

  v8f acc = {};
#pragma unroll 4
  for (int k = 0; k < IN_D; k += 4) {
    v2f ax = *(const v2f*)(xr + k);         // A frag of x
    v2f ag = *(const v2f*)(gr + k);         // A frag of agg
    ag = ag * rdeg;                         // fused mean-normalization
    const int brow = (k + koff) * OUT_D + n0 + lr;
    v2f bsf, bnf;
    bsf.x = Ws[brow];
    bsf.y = Ws[brow + OUT_D];
    bnf.x = Wn[brow];
    bnf.y = Wn[brow + OUT_D];
    // D = A*B + C, exact fp32 matrix op (V_WMMA_F32_16X16X4_F32)
    acc = __builtin_amdgcn_wmma_f32_16x16x4_f32(false, ax, false, bsf,
                                                (short)0, acc, false, false);
    acc = __builtin_amdgcn_wmma_f32_16x16x4_f32(false, ag, false, bnf,
                                                (short)0, acc, false, false);
  }

  const float bias = bsv[n0 + lr] + bnv[n0 + lr];
  float* op = out + (long)(m0 + hi * 8) * OUT_D + n0 + lr;

  // Block-uniform full-tile test: hot path has zero exec-mask churn.
  if (m0 + 16 <= nNodes) {
#pragma unroll
    for (int r = 0; r < 8; ++r) {
      op[(long)r * OUT_D] = fmaxf(acc[r] + bias, 0.0f);
    }
  } else {
#pragma unroll
    for (int r = 0; r < 8; ++r) {
      const int row = m0 + hi * 8 + r;      // C/D layout: lanes16-31 hold M=r+8
      if (row < nNodes) {
        op[(long)r * OUT_D] = fmaxf(acc[r] + bias, 0.0f);
      }
    }
  }
}

extern "C" void kernel_launch(void* const* d_in, const int* in_sizes, int n_in,
                              void* d_out, int out_size, void* d_ws, size_t ws_size,
                              hipStream_t stream) {
  const float*     x  = (const float*)d_in[0];
  const long long* ei = (const long long*)d_in[1];
  const float*     Ws = (const float*)d_in[2];
  const float*     bs = (const float*)d_in[3];
  const float*     Wn = (const float*)d_in[4];
  const float*     bn = (const float*)d_in[5];
  float* out = (float*)d_out;

  const int N = in_sizes[0] / IN_D;
  const int E = in_sizes[1] / 2;

  float* agg = (float*)d_ws;                 // N * 128 floats
  float* deg = agg + (long)N * IN_D;         // N floats

  const long zn = (long)N * IN_D + N;
  gnn_zero_kernel<<<2048, 512, 0, stream>>>(agg, zn);

  const long scatterThreads = (long)E * 32;
  const int scatterBlocks = (int)((scatterThreads + 255) / 256);
  gnn_scatter_kernel<<<scatterBlocks, 256, 0, stream>>>(x, ei, agg, deg, E);

  const int mTiles = (N + 15) / 16;          // 6250 for N=100000
  gnn_gemm_kernel<<<mTiles, 256, 0, stream>>>(x, agg, deg, Ws, bs, Wn, bn, out, N);
}